// CustomNetwork_31585189494999
// MI455X (gfx1250) — compile-verified
//
#include <hip/hip_runtime.h>

#define B_TOT 65536
#define NTOT  320
#define NSUB  256

typedef float v2f __attribute__((ext_vector_type(2)));
typedef float v8f __attribute__((ext_vector_type(8)));

__device__ __forceinline__ v8f wmma4(v2f a, v2f b, v8f c) {
  // D = A(16x4 f32) * B(4x16 f32) + C(16x16 f32)
  return __builtin_amdgcn_wmma_f32_16x16x4_f32(false, a, false, b, (short)0, c,
                                               false, false);
}

__device__ __forceinline__ float selu_f(float x) {
  const float l  = 1.0507009873554805f;   // lambda
  const float la = 1.7580993408473766f;   // lambda * alpha
  return x > 0.f ? l * x : la * (__expf(x) - 1.f);
}

__global__ void zero_ws(float* ws) {
  int i = blockIdx.x * blockDim.x + threadIdx.x;
  if (i < 16384) ws[i] = 0.f;
}

// mean/var -> (scale, shift) so norm is h*scale + shift
__global__ void finalize_stats(const float* __restrict__ sum,
                               const float* __restrict__ sumsq,
                               const float* __restrict__ gamma,
                               const float* __restrict__ beta,
                               float* __restrict__ scale,
                               float* __restrict__ shift) {
  int i = blockIdx.x * blockDim.x + threadIdx.x;
  if (i < NSUB * 8) {
    const float inv = 1.f / (float)B_TOT;
    float m  = sum[i] * inv;
    float v  = sumsq[i] * inv - m * m;
    float r  = rsqrtf(v + 1e-5f);
    float sc = r * gamma[i];
    scale[i] = sc;
    shift[i] = beta[i] - m * sc;
  }
}

// PASS 1: stats of h1.  PASS 2: stats of h2.  PASS 3: final output.
template <int PASS>
__global__ __launch_bounds__(256) void subnet_pass(
    const float* __restrict__ x,  const int*   __restrict__ pidx,
    const float* __restrict__ W1, const float* __restrict__ b1,
    const float* __restrict__ W2, const float* __restrict__ b2,
    const float* __restrict__ W3, const float* __restrict__ b3,
    const float* __restrict__ sc1, const float* __restrict__ sh1,
    const float* __restrict__ sc2, const float* __restrict__ sh2,
    float* __restrict__ outSum, float* __restrict__ outSumsq,
    float* __restrict__ out) {
  __shared__ float lds[8][16 * 17];
  const int n    = blockIdx.x;
  const int w    = threadIdx.x >> 5;
  const int lane = threadIdx.x & 31;
  const int col  = lane & 15;        // N column (layer out-channel), valid < 8
  const int hgrp = lane >> 4;        // 0: K=0,1 / rows 0-7 ; 1: K=2,3 / rows 8-15
  float* tl = &lds[w][0];

  int pi[8];
#pragma unroll
  for (int k = 0; k < 8; ++k) pi[k] = pidx[n * 8 + k];

  // Layer-1 B operands: B[K=d][N=h], K = v + 2*hgrp (+4 for hi op)
  v2f B1lo, B1hi;
#pragma unroll
  for (int v = 0; v < 2; ++v) {
    int kr = v + 2 * hgrp;
    B1lo[v] = (col < 8) ? W1[n * 64 + kr * 8 + col]       : 0.f;
    B1hi[v] = (col < 8) ? W1[n * 64 + (kr + 4) * 8 + col] : 0.f;
  }
  float bias1 = (col < 8) ? b1[n * 8 + col] : 0.f;

  v2f B2lo = {}, B2hi = {};
  float bias2 = 0.f, s1v = 0.f, t1v = 0.f, s2v = 0.f, t2v = 0.f;
  float w3r[8];
  float b3n = 0.f;
  if (PASS >= 2) {
#pragma unroll
    for (int v = 0; v < 2; ++v) {
      int kr = v + 2 * hgrp;
      B2lo[v] = (col < 8) ? W2[n * 64 + kr * 8 + col]       : 0.f;
      B2hi[v] = (col < 8) ? W2[n * 64 + (kr + 4) * 8 + col] : 0.f;
    }
    bias2 = (col < 8) ? b2[n * 8 + col] : 0.f;
    s1v   = (col < 8) ? sc1[n * 8 + col] : 0.f;
    t1v   = (col < 8) ? sh1[n * 8 + col] : 0.f;
  }
  if (PASS == 3) {
    s2v = (col < 8) ? sc2[n * 8 + col] : 0.f;
    t2v = (col < 8) ? sh2[n * 8 + col] : 0.f;
#pragma unroll
    for (int h = 0; h < 8; ++h) w3r[h] = W3[n * 8 + h];
    b3n = b3[n];
  }

  float accS = 0.f, accQ = 0.f;

  const int T = 16;                        // 16-row tiles per wave
  const int tile0 = (blockIdx.y * 8 + w) * T;
  for (int i = 0; i < T; ++i) {
    const int b0 = (tile0 + i) * 16;
    // A operand: A[M=col][K=d], gathered from L2-resident x
    const float* xr = x + (size_t)(b0 + col) * NTOT;
    v2f Alo, Ahi;
#pragma unroll
    for (int v = 0; v < 2; ++v) {
      int d  = v + 2 * hgrp;
      Alo[v] = xr[pi[d]];
      Ahi[v] = xr[pi[d + 4]];
    }
    v8f c;
#pragma unroll
    for (int r = 0; r < 8; ++r) c[r] = bias1;
    c = wmma4(Alo, B1lo, c);
    c = wmma4(Ahi, B1hi, c);

    if (PASS == 1) {
      // per-lane partial column sums; halves folded once after the loop
#pragma unroll
      for (int r = 0; r < 8; ++r) { accS += c[r]; accQ += c[r] * c[r]; }
    } else {
      // norm1 + SELU in C layout, transpose through LDS into A layout
#pragma unroll
      for (int r = 0; r < 8; ++r) {
        float hv = selu_f(c[r] * s1v + t1v);
        if (col < 8) tl[(r + 8 * hgrp) * 17 + col] = hv;
      }
      asm volatile("s_wait_dscnt 0" ::: "memory");
      v2f A2lo, A2hi;
#pragma unroll
      for (int v = 0; v < 2; ++v) {
        int d   = v + 2 * hgrp;
        A2lo[v] = tl[col * 17 + d];
        A2hi[v] = tl[col * 17 + d + 4];
      }
      v8f c2;
#pragma unroll
      for (int r = 0; r < 8; ++r) c2[r] = bias2;
      c2 = wmma4(A2lo, B2lo, c2);
      c2 = wmma4(A2hi, B2hi, c2);

      if (PASS == 2) {
#pragma unroll
        for (int r = 0; r < 8; ++r) { accS += c2[r]; accQ += c2[r] * c2[r]; }
      } else {
        // norm2 + SELU, then 8->1 projection via LDS row dot
#pragma unroll
        for (int r = 0; r < 8; ++r) {
          float hv = selu_f(c2[r] * s2v + t2v);
          if (col < 8) tl[(r + 8 * hgrp) * 17 + col] = hv;
        }
        asm volatile("s_wait_dscnt 0" ::: "memory");
        if (lane < 16) {
          float s = b3n;
#pragma unroll
          for (int h = 0; h < 8; ++h) s += tl[lane * 17 + h] * w3r[h];
          if (n == NSUB - 1) s = 1.f / (1.f + __expf(-s));
          // out is write-once / never re-read: NT store keeps x resident in L2
          __builtin_nontemporal_store(s, &out[(size_t)(b0 + lane) * NSUB + n]);
        }
      }
    }
  }

  if (PASS == 1 || PASS == 2) {
    // fold lane halves (rows 0-7 vs 8-15 of each column) once per wave
    accS += __shfl_xor(accS, 16, 32);
    accQ += __shfl_xor(accQ, 16, 32);
    if (lane < 8) {
      atomicAdd(&outSum[n * 8 + lane], accS);
      atomicAdd(&outSumsq[n * 8 + lane], accQ);
    }
  }
}

extern "C" void kernel_launch(void* const* d_in, const int* in_sizes, int n_in,
                              void* d_out, int out_size, void* d_ws, size_t ws_size,
                              hipStream_t stream) {
  const float* x   = (const float*)d_in[0];
  const int*  pidx = (const int*)  d_in[1];
  const float* W1  = (const float*)d_in[2];
  const float* b1  = (const float*)d_in[3];
  const float* g1  = (const float*)d_in[4];
  const float* be1 = (const float*)d_in[5];
  const float* W2  = (const float*)d_in[6];
  const float* b2  = (const float*)d_in[7];
  const float* g2  = (const float*)d_in[8];
  const float* be2 = (const float*)d_in[9];
  const float* W3  = (const float*)d_in[10];
  const float* b3  = (const float*)d_in[11];
  float* out = (float*)d_out;
  float* ws  = (float*)d_ws;

  float* sum1 = ws;          float* sq1 = ws + 2048;
  float* sc1  = ws + 4096;   float* sh1 = ws + 6144;
  float* sum2 = ws + 8192;   float* sq2 = ws + 10240;
  float* sc2  = ws + 12288;  float* sh2 = ws + 14336;

  zero_ws<<<64, 256, 0, stream>>>(ws);

  dim3 grid(NSUB, 32);  // 256 subnets x 32 batch chunks; 8 waves x 16 tiles each
  subnet_pass<1><<<grid, 256, 0, stream>>>(x, pidx, W1, b1, W2, b2, W3, b3,
                                           sc1, sh1, sc2, sh2, sum1, sq1, out);
  finalize_stats<<<8, 256, 0, stream>>>(sum1, sq1, g1, be1, sc1, sh1);
  subnet_pass<2><<<grid, 256, 0, stream>>>(x, pidx, W1, b1, W2, b2, W3, b3,
                                           sc1, sh1, sc2, sh2, sum2, sq2, out);
  finalize_stats<<<8, 256, 0, stream>>>(sum2, sq2, g2, be2, sc2, sh2);
  subnet_pass<3><<<grid, 256, 0, stream>>>(x, pidx, W1, b1, W2, b2, W3, b3,
                                           sc1, sh1, sc2, sh2, sum2, sq2, out);
}